// NGCF_70403103916221
// MI455X (gfx1250) — compile-verified
//
#include <hip/hip_runtime.h>

#define NUM_USER 100000
#define NUM_ITEM 50000
#define NN (NUM_USER + NUM_ITEM)   // 150000, divisible by 16
#define DD 64
#define LL 3

typedef __attribute__((ext_vector_type(2))) float v2f;
typedef __attribute__((ext_vector_type(8))) float v8f;

// ---------------- utility kernels ----------------

__global__ void ngcf_zero_f32(float* __restrict__ p, int n) {
  int i = blockIdx.x * blockDim.x + threadIdx.x;
  int stride = gridDim.x * blockDim.x;
  for (; i < n; i += stride) p[i] = 0.0f;
}

__global__ void ngcf_deg(const int* __restrict__ ui, const int* __restrict__ ii,
                         float* __restrict__ deg, int E) {
  int e = blockIdx.x * blockDim.x + threadIdx.x;
  if (e >= E) return;
  atomicAdd(&deg[ui[e]], 1.0f);
  atomicAdd(&deg[NUM_USER + ii[e]], 1.0f);
}

__global__ void ngcf_dinvsqrt(const float* __restrict__ deg, float* __restrict__ dis, int n) {
  int i = blockIdx.x * blockDim.x + threadIdx.x;
  if (i < n) dis[i] = rsqrtf(deg[i] + 1e-10f);
}

__global__ void ngcf_init(const float* __restrict__ ue, const float* __restrict__ ie,
                          float* __restrict__ h, float* __restrict__ acc) {
  int i = blockIdx.x * blockDim.x + threadIdx.x;
  if (i >= NN * DD) return;
  float v = (i < NUM_USER * DD) ? ue[i] : ie[i - NUM_USER * DD];
  h[i] = v;
  acc[i] = v;
}

// ---------------- SpMM: one wave32 per edge, float2 per lane ----------------
// agg[u] += val * h[v];  agg[v] += val * h[u]   (L2-resident scatter-add)

__global__ void ngcf_spmm(const int* __restrict__ ui, const int* __restrict__ ii,
                          const float* __restrict__ dis, const float* __restrict__ h,
                          float* __restrict__ agg, int E) {
  int t = blockIdx.x * blockDim.x + threadIdx.x;
  int e = t >> 5;
  int lane = t & 31;
  if (e >= E) return;
  int u = ui[e];
  int v = NUM_USER + ii[e];
  float val = dis[u] * dis[v];
  size_t uo = (size_t)u * DD + (size_t)lane * 2;
  size_t vo = (size_t)v * DD + (size_t)lane * 2;
  float2 hu = *(const float2*)(h + uo);
  float2 hv = *(const float2*)(h + vo);
  atomicAdd(&agg[uo + 0], val * hv.x);
  atomicAdd(&agg[uo + 1], val * hv.y);
  atomicAdd(&agg[vo + 0], val * hu.x);
  atomicAdd(&agg[vo + 1], val * hu.y);
}

// ---------------- Dense GEMM via f32 WMMA: h = relu(agg @ W^T); acc += h ----
// Out[m,n] = sum_k agg[m,k] * W[n,k].  Tile 16x16, K=64 in 16 steps of 4.
// Block = 128 threads (4 waves); wave w owns the 16-column tile n0 = 16*w.

__global__ __launch_bounds__(128) void ngcf_gemm_relu_acc(
    const float* __restrict__ agg, const float* __restrict__ Wl,
    float* __restrict__ h, float* __restrict__ acc) {
  unsigned lane = threadIdx.x & 31u;
  unsigned wave = threadIdx.x >> 5;
  unsigned m0 = blockIdx.x * 16u;
  unsigned n0 = wave * 16u;
  unsigned lh = lane & 15u;        // M (for A) / N (for B) within tile
  unsigned khalf = lane >> 4;      // 0: K={0,1}, 1: K={2,3} per 4-step
  unsigned koff = khalf * 2u;

  const float* arow = agg + (size_t)(m0 + lh) * DD + koff;  // A: agg[m, k]
  const float* brow = Wl  + (size_t)(n0 + lh) * DD + koff;  // B: W[n, k] = (W^T)[k, n]

  v8f c = {0.f, 0.f, 0.f, 0.f, 0.f, 0.f, 0.f, 0.f};
#pragma unroll
  for (int kk = 0; kk < DD; kk += 4) {
    float2 av = *(const float2*)(arow + kk);
    float2 bv = *(const float2*)(brow + kk);
    v2f a; a.x = av.x; a.y = av.y;
    v2f b; b.x = bv.x; b.y = bv.y;
    // D = A(16x4,f32) x B(4x16,f32) + C(16x16,f32)
    c = __builtin_amdgcn_wmma_f32_16x16x4_f32(
        /*neg_a=*/false, a, /*neg_b=*/false, b,
        /*c_mod=*/(short)0, c, /*reuse_a=*/false, /*reuse_b=*/false);
  }

  // C/D layout: VGPR r -> M = r + (lane<16 ? 0 : 8), N = lane%16
  unsigned mbase = m0 + khalf * 8u;
#pragma unroll
  for (int r = 0; r < 8; r++) {
    float v = fmaxf(c[r], 0.0f);
    size_t o = (size_t)(mbase + r) * DD + n0 + lh;
    h[o] = v;
    acc[o] += v;
  }
}

__global__ void ngcf_scale(float* __restrict__ acc, int n) {
  int i = blockIdx.x * blockDim.x + threadIdx.x;
  if (i < n) acc[i] *= 0.25f;  // 1 / (L+1)
}

// ---------------- launch ----------------

extern "C" void kernel_launch(void* const* d_in, const int* in_sizes, int n_in,
                              void* d_out, int out_size, void* d_ws, size_t ws_size,
                              hipStream_t stream) {
  const float* ue = (const float*)d_in[0];   // [NUM_USER, 64]
  const float* ie = (const float*)d_in[1];   // [NUM_ITEM, 64]
  const float* W  = (const float*)d_in[2];   // [3, 64, 64]
  const int*   ui = (const int*)d_in[3];     // [E]
  const int*   ii = (const int*)d_in[4];     // [E]
  const int E = in_sizes[3];

  float* acc = (float*)d_out;                // [N, 64] accumulator -> output

  float* deg = (float*)d_ws;                 // N
  float* dis = deg + NN;                     // N
  float* h   = dis + NN;                     // N*64
  float* agg = h + (size_t)NN * DD;          // N*64

  const int feat = NN * DD;

  ngcf_zero_f32<<<(NN + 255) / 256, 256, 0, stream>>>(deg, NN);
  ngcf_deg<<<(E + 255) / 256, 256, 0, stream>>>(ui, ii, deg, E);
  ngcf_dinvsqrt<<<(NN + 255) / 256, 256, 0, stream>>>(deg, dis, NN);
  ngcf_init<<<(feat + 255) / 256, 256, 0, stream>>>(ue, ie, h, acc);

  const int spmm_blocks = (int)(((size_t)E * 32 + 255) / 256);
  for (int l = 0; l < LL; ++l) {
    ngcf_zero_f32<<<4096, 256, 0, stream>>>(agg, feat);
    ngcf_spmm<<<spmm_blocks, 256, 0, stream>>>(ui, ii, dis, h, agg, E);
    ngcf_gemm_relu_acc<<<NN / 16, 128, 0, stream>>>(agg, W + (size_t)l * DD * DD, h, acc);
  }

  ngcf_scale<<<(feat + 255) / 256, 256, 0, stream>>>(acc, feat);
}